// ChirpLinker_39367670235652
// MI455X (gfx1250) — compile-verified
//
#include <hip/hip_runtime.h>
#include <stdint.h>

// Problem constants (from reference: tokens (16, 512, 64, 9) fp32)
#define B_ 16
#define W_ 512
#define K_ 64
#define C_ 9
#define NELEM_ (B_ * W_ * K_)

// ---------------------------------------------------------------------------
// CDNA5 feature gates (device pass only; host pass takes fallbacks)
// ---------------------------------------------------------------------------
#if defined(__AMDGCN__) && __has_builtin(__builtin_amdgcn_global_load_async_to_lds_b32) && __has_builtin(__builtin_amdgcn_s_wait_asynccnt)
#define USE_ASYNC_LDS 1
#else
#define USE_ASYNC_LDS 0
#endif

#if defined(__AMDGCN__) && __has_builtin(__builtin_amdgcn_s_wait_tensorcnt)
#define HAVE_TENSORCNT 1
#else
#define HAVE_TENSORCNT 0
#endif

__device__ __forceinline__ unsigned ballot32(bool p) {
#if defined(__AMDGCN__) && __has_builtin(__builtin_amdgcn_ballot_w32)
  return __builtin_amdgcn_ballot_w32(p);
#else
  return (unsigned)__ballot(p);  // wave32: low 32 bits are the wave's mask
#endif
}

#if USE_ASYNC_LDS
typedef __attribute__((address_space(1))) int gint_t;  // prints as "__device__ int"
typedef __attribute__((address_space(3))) int lint_t;  // prints as "__shared__ int"
__device__ __forceinline__ void async_ld_f32(const float* g, float* l) {
  // global_load_async_to_lds_b32: (global src, LDS dst, imm offset, imm cpol)
  __builtin_amdgcn_global_load_async_to_lds_b32((gint_t*)(float*)g, (lint_t*)l, 0, 0);
}
#endif

// wrap(x) = (x + pi) mod 2pi - pi with jnp.mod semantics (result sign of divisor)
__device__ __forceinline__ float wrapf(float x) {
  const float PI_F  = 3.14159265358979323846f;
  const float TPI_F = 6.28318530717958647692f;
  float m = fmodf(x + PI_F, TPI_F);
  if (m < 0.0f) m += TPI_F;
  return m - PI_F;
}

// ---------------------------------------------------------------------------
// Kernel 1: per-transition matcher.  One 64-thread block per (b, w).
// Packs nxt[5:0] | prv[11:6] | has_next[12] into link[]; writes has_prev[w+1].
// w == W-1 block zeroes link[b, W-1, :] and has_prev[b, 0, :].
// ---------------------------------------------------------------------------
__global__ void __launch_bounds__(K_) link_kernel(const float* __restrict__ tok,
                                                  int* __restrict__ link,
                                                  unsigned char* __restrict__ has_prev) {
#if HAVE_TENSORCNT
  __builtin_amdgcn_s_wait_tensorcnt(0);  // CDNA5 split-counter wait (TENSORcnt==0, no-op)
#endif
  const int bw = blockIdx.x;
  const int b = bw / W_, w = bw % W_;
  const int k = threadIdx.x;
  if (w == W_ - 1) {
    link[((size_t)b * W_ + w) * K_ + k] = 0;        // nxt_f / has_next_f zero pad
    has_prev[((size_t)b * W_) * K_ + k] = 0;        // has_prev[:,0,:] = false
    return;
  }
  __shared__ float fe[K_], Ae[K_], pe[K_], fn[K_], An[K_], pn[K_];
  __shared__ unsigned char nxs[K_], pvs[K_], hns[K_];

  const float* te = tok + (((size_t)b * W_ + w) * K_ + k) * C_;
  const float* tn = tok + (((size_t)b * W_ + w + 1) * K_ + k) * C_;
#if USE_ASYNC_LDS
  async_ld_f32(te + 4, &fe[k]);
  async_ld_f32(te + 6, &Ae[k]);
  async_ld_f32(te + 8, &pe[k]);
  async_ld_f32(tn + 3, &fn[k]);
  async_ld_f32(tn + 5, &An[k]);
  async_ld_f32(tn + 7, &pn[k]);
  __builtin_amdgcn_s_wait_asynccnt(0);
#else
  fe[k] = te[4]; Ae[k] = te[6]; pe[k] = te[8];
  fn[k] = tn[3]; An[k] = tn[5]; pn[k] = tn[7];
#endif
  __syncthreads();

  // Row argmin (i = k): cost = valid ? df : inf; strict < keeps first index (jnp.argmin).
  const float fei = fe[k], Aei = Ae[k], pei = pe[k];
  float best = __builtin_inff();
  int bi = 0;
  bool any = false;
  for (int j = 0; j < K_; ++j) {
    float df = fabsf(fei - fn[j]);
    float dA = fabsf(Aei - An[j]);
    float dp = fabsf(wrapf(pn[j] - pei));
    bool v = (df <= 0.5f) && (dA <= 0.5f) && (dp <= 1.0f);
    float c = v ? df : __builtin_inff();
    if (c < best) { best = c; bi = j; }
    any |= v;
  }
  nxs[k] = (unsigned char)bi;

  // Column argmin (j = k)
  const float fnj = fn[k], Anj = An[k], pnj = pn[k];
  float bestc = __builtin_inff();
  int bj = 0;
  for (int i = 0; i < K_; ++i) {
    float df = fabsf(fe[i] - fnj);
    float dA = fabsf(Ae[i] - Anj);
    float dp = fabsf(wrapf(pnj - pe[i]));
    bool v = (df <= 0.5f) && (dA <= 0.5f) && (dp <= 1.0f);
    float c = v ? df : __builtin_inff();
    if (c < bestc) { bestc = c; bj = i; }
  }
  pvs[k] = (unsigned char)bj;
  __syncthreads();

  const bool hn = any && (pvs[nxs[k]] == (unsigned char)k);  // mutual match
  hns[k] = hn ? 1 : 0;
  link[((size_t)b * W_ + w) * K_ + k] =
      (int)nxs[k] | ((int)pvs[k] << 6) | (hn ? (1 << 12) : 0);
  __syncthreads();

  const int pv = pvs[k];
  has_prev[((size_t)b * W_ + (w + 1)) * K_ + k] =
      (hns[pv] && nxs[pv] == (unsigned char)k) ? 1 : 0;
}

// ---------------------------------------------------------------------------
// Kernel 2: backward scan over W.  One 64-thread block per batch.
// Produces suffix_snr, qualify, and per-(b,w) qualify counts (wave32 ballots).
// ---------------------------------------------------------------------------
__global__ void __launch_bounds__(K_) bwd_kernel(const float* __restrict__ tok,
                                                 const int* __restrict__ link,
                                                 const unsigned char* __restrict__ has_prev,
                                                 unsigned char* __restrict__ qualify,
                                                 float* __restrict__ suffix_snr,
                                                 int* __restrict__ row_count) {
  const int b = blockIdx.x, k = threadIdx.x;
  __shared__ int   lenN[K_];
  __shared__ float snrN[K_];
  __shared__ int   wcnt[2];
  for (int w = W_ - 1; w >= 0; --w) {
    size_t e = ((size_t)b * W_ + w) * K_ + k;
    if (w > 0)  // gfx1250 global_prefetch_b8 of the next iteration's token
      __builtin_prefetch((const void*)(tok + (e - (size_t)K_) * C_), 0, 3);
    int L = link[e];
    int nx = L & 63;
    bool hn = (L >> 12) & 1;             // w==W-1 => link==0 => hn false (no LDS dep)
    float s0 = tok[e * C_];
    float ssq = s0 * s0;
    int   ln = 1 + (hn ? lenN[nx] : 0);
    float sn = ssq + (hn ? snrN[nx] : 0.0f);
    bool q = (!has_prev[e]) && (ln >= 2);  // MIN_LENGTH = 2
    qualify[e] = q ? 1 : 0;
    suffix_snr[e] = sn;
    unsigned m = ballot32(q);
    __syncthreads();                       // all gathers of step w done
    lenN[k] = ln; snrN[k] = sn;
    if ((k & 31) == 0) wcnt[k >> 5] = __popc(m);
    __syncthreads();                       // carry + counts visible
    if (k == 0) row_count[b * W_ + w] = wcnt[0] + wcnt[1];
  }
}

// ---------------------------------------------------------------------------
// Kernel 3: exclusive prefix sum of 8192 row counts (global cumsum of qualify).
// Single block, 1024 threads, 8 rows/thread, Hillis-Steele block scan.
// ---------------------------------------------------------------------------
__global__ void __launch_bounds__(1024) scan_kernel(int* __restrict__ rows) {
  __shared__ int part[1024];
  const int t = threadIdx.x;
  const int base = t * 8;
  int v[8];
  int s = 0;
  for (int i = 0; i < 8; ++i) { v[i] = rows[base + i]; s += v[i]; }
  part[t] = s;
  __syncthreads();
  for (int off = 1; off < 1024; off <<= 1) {
    int add = (t >= off) ? part[t - off] : 0;
    __syncthreads();
    part[t] += add;
    __syncthreads();
  }
  int excl = part[t] - s;  // exclusive prefix for this thread's first row
  for (int i = 0; i < 8; ++i) { rows[base + i] = excl; excl += v[i]; }
}

// ---------------------------------------------------------------------------
// Kernel 4: forward scan over W.  One 64-thread block per batch.
// head_val via ballot prefix + global row offsets; injective conflict-free
// LDS scatter for the carry (mutual matching => nxt unique among has_next).
// ---------------------------------------------------------------------------
__global__ void __launch_bounds__(K_) fwd_kernel(const int* __restrict__ link,
                                                 const unsigned char* __restrict__ has_prev,
                                                 const unsigned char* __restrict__ qualify,
                                                 const float* __restrict__ suffix_snr,
                                                 const int* __restrict__ row_off,
                                                 int* __restrict__ chain_id,
                                                 float* __restrict__ tot_snr) {
  const int b = blockIdx.x, k = threadIdx.x;
  __shared__ int   cidA[K_]; __shared__ float csnA[K_];   // carry in
  __shared__ int   cidB[K_]; __shared__ float csnB[K_];   // carry out (scatter)
  __shared__ int   w0cnt;
  cidA[k] = -1; csnA[k] = 0.0f;
  __syncthreads();
  for (int w = 0; w < W_; ++w) {
    size_t e = ((size_t)b * W_ + w) * K_ + k;
    int L = link[e];
    int nx = L & 63;
    bool hn = (L >> 12) & 1;
    bool hp = has_prev[e] != 0;
    bool q  = qualify[e] != 0;
    float ssn = suffix_snr[e];
    unsigned m = ballot32(q);
    if (k == 0) w0cnt = __popc(m);  // wave0's qualify count
    __syncthreads();
    int lane = k & 31;
    int incl = __popc(m & (0xFFFFFFFFu >> (31 - lane)));  // inclusive in-wave prefix
    int before = row_off[b * W_ + w] + ((k >= 32) ? w0cnt : 0) + incl;
    int hv = q ? (before - 1) : -1;       // head_val (global cumsum - 1)
    int   cid = hp ? cidA[k] : hv;
    float csn = hp ? csnA[k] : ssn;
    chain_id[e] = cid;
    tot_snr[e]  = csn;
    cidB[k] = 0; csnB[k] = 0.0f;
    __syncthreads();
    if (hn) { cidB[nx] = cid + 1; csnB[nx] = csn; }  // injective: no conflicts
    __syncthreads();
    cidA[k] = cidB[k] - 1;   // zeros.at[nx].add(...) - 1
    csnA[k] = csnB[k];
  }
}

// ---------------------------------------------------------------------------
// Kernel 5: enrichment.  One 64-thread block per (b, w).
// Incoming edge resolved through prv (inverse map), <=1 per target.
// ---------------------------------------------------------------------------
__global__ void __launch_bounds__(K_) finalize_kernel(const float* __restrict__ tok,
                                                      const int* __restrict__ link,
                                                      const unsigned char* __restrict__ has_prev,
                                                      const int* __restrict__ chain_id,
                                                      const float* __restrict__ tot_snr,
                                                      float* __restrict__ out) {
  const int bw = blockIdx.x;
  const int b = bw / W_, w = bw % W_;
  const int k = threadIdx.x;
  const size_t e = ((size_t)b * W_ + w) * K_ + k;
  const float* t = tok + e * C_;
  float t0 = t[0], t1 = t[1], t2 = t[2], t3 = t[3], t4 = t[4];
  float t5 = t[5], t6 = t[6], t7 = t[7], t8 = t[8];

  int cid = chain_id[e];
  bool assigned = cid >= 0;
  float ts = tot_snr[e];
  float comb = sqrtf(ts > 0.0f ? ts : 1.0f);
  float col0 = assigned ? comb : t0;

  // Outgoing edge: col4/col6/col8 (last window keeps raw values)
  float col4 = t4, col6 = t6, col8 = t8;
  if (w < W_ - 1) {
    int L = link[e];
    if (((L >> 12) & 1) && assigned) {
      int nx = L & 63;
      const float* tt = tok + (((size_t)b * W_ + w + 1) * K_ + nx) * C_;
      float fn_ = tt[3], An_ = tt[5], pn_ = tt[7];
      col4 = 0.5f * (t4 + fn_);
      col6 = 0.5f * (t6 + An_);
      col8 = t8 + 0.5f * wrapf(pn_ - t8);
    }
  }

  // Incoming edge: col3/col5/col7 (first window keeps raw values)
  float col3 = t3, col5 = t5, col7 = t7;
  if (w > 0 && has_prev[e]) {
    size_t ep = ((size_t)b * W_ + (w - 1)) * K_ + k;
    int pv = (link[ep] >> 6) & 63;               // mutual source index
    size_t es = ((size_t)b * W_ + (w - 1)) * K_ + pv;
    if (chain_id[es] >= 0) {                     // edge = has_next & assigned(src)
      const float* tp = tok + es * C_;
      float fep = tp[4], Aep = tp[6], pep = tp[8];
      float favg = 0.5f * (fep + t3);            // gathered f_n == t3 (nxt[pv]==k)
      float Aavg = 0.5f * (Aep + t5);
      float hc = 0.5f * wrapf(t7 - pep);
      col3 = t3 + (favg - t3);                   // base + delta (reference FP order)
      col5 = t5 + (Aavg - t5);
      col7 = t7 + ((t7 - hc) - t7);
    }
  }

  float* o = out + e * 10;
  o[0] = col0; o[1] = t1; o[2] = t2; o[3] = col3; o[4] = col4;
  o[5] = col5; o[6] = col6; o[7] = col7; o[8] = col8; o[9] = (float)cid;
}

// ---------------------------------------------------------------------------
// Launch
// ---------------------------------------------------------------------------
extern "C" void kernel_launch(void* const* d_in, const int* in_sizes, int n_in,
                              void* d_out, int out_size, void* d_ws, size_t ws_size,
                              hipStream_t stream) {
  (void)in_sizes; (void)n_in; (void)out_size; (void)ws_size;
  const float* tok = (const float*)d_in[0];
  float* out = (float*)d_out;

  uintptr_t p = (uintptr_t)d_ws;
  auto carve = [&p](size_t bytes) -> void* {
    p = (p + 255) & ~(uintptr_t)255;
    void* r = (void*)p;
    p += bytes;
    return r;
  };
  int*           link   = (int*)carve(sizeof(int) * NELEM_);
  unsigned char* hasp   = (unsigned char*)carve(NELEM_);
  unsigned char* qual   = (unsigned char*)carve(NELEM_);
  float*         ssnr   = (float*)carve(sizeof(float) * NELEM_);
  int*           chain  = (int*)carve(sizeof(int) * NELEM_);
  float*         tsnr   = (float*)carve(sizeof(float) * NELEM_);
  int*           rows   = (int*)carve(sizeof(int) * B_ * W_);

  link_kernel<<<dim3(B_ * W_), dim3(K_), 0, stream>>>(tok, link, hasp);
  bwd_kernel<<<dim3(B_), dim3(K_), 0, stream>>>(tok, link, hasp, qual, ssnr, rows);
  scan_kernel<<<dim3(1), dim3(1024), 0, stream>>>(rows);
  fwd_kernel<<<dim3(B_), dim3(K_), 0, stream>>>(link, hasp, qual, ssnr, rows, chain, tsnr);
  finalize_kernel<<<dim3(B_ * W_), dim3(K_), 0, stream>>>(tok, link, hasp, chain, tsnr, out);
}